// PrefillAttnModel_7670811591161
// MI455X (gfx1250) — compile-verified
//
#include <hip/hip_runtime.h>
#include <hip/hip_bf16.h>
#include <cstddef>

// ---------------------------------------------------------------------------
// MLA prefill on gfx1250 (MI455X).  All matrix math on V_WMMA_F32_16X16X4_F32
// (native CDNA5 fp32 WMMA).  Fragment operands laid out in LDS so every
// WMMA operand is a single aligned ds_load_b64 (no repacking moves), with
// bank-conflict-free strides and chunked load->MMA scheduling.
// ---------------------------------------------------------------------------

typedef float v2f __attribute__((ext_vector_type(2)));
typedef float v8f __attribute__((ext_vector_type(8)));

#define S_LEN   2048
#define HID_D   2048
#define NH      16
#define D_NOPE  128
#define D_ROPE  64
#define D_V     128
#define Q_HEAD  192
#define QL_D    1536
#define KVL_D   512
#define CKV_W   576     // KVL + D_ROPE
#define KV_W    4096    // NH*(D_NOPE+D_V)
#define QW_D    3072    // NH*Q_HEAD
#define ATT_SCALE 0.07216878364870322f   // 192^-0.5
#define RMS_EPS 1e-6f

static __device__ __forceinline__ v8f wmma4(v2f a, v2f b, v8f c) {
  // D = A(16x4,f32) * B(4x16,f32) + C(16x16,f32)
  return __builtin_amdgcn_wmma_f32_16x16x4_f32(false, a, false, b,
                                               (short)0, c, false, false);
}

// ---------------------------------------------------------------------------
// Generic fp32 WMMA GEMM:  C[M,N] = A[M,K] @ B[K,N]
// Workgroup tile 128x64, 8 waves each owning a 16-row strip (4 n-tiles).
// Requires M%128==0, N%64==0, K%16==0 (true for all GEMMs here).
// LDS layouts:
//   lda: row-major, stride 20 (even -> aligned v2f frags; 20*l mod 64 distinct)
//   ldb: k-pair interleaved [pair][col][2], pair stride 160 words
//        (+160 -> +32 banks, so half-waves hit disjoint bank sets)
// ---------------------------------------------------------------------------
#define GBM 128
#define GBN 64
#define GBK 16
#define LDA_STR 20
#define LDB_PSTR 160

__global__ __launch_bounds__(256) void gemm_f32_wmma(
    const float* __restrict__ A, const float* __restrict__ B,
    float* __restrict__ C, int M, int N, int K)
{
  __shared__ float lda[GBM * LDA_STR];     // 10 KB
  __shared__ float ldb[8 * LDB_PSTR];      //  5 KB
  const int t    = threadIdx.x;
  const int wave = t >> 5;
  const int lane = t & 31;
  const int lh   = lane & 15;
  const int hs   = lane >> 4;              // half-wave: K offset +2
  const int mblk = blockIdx.y * GBM;
  const int nblk = blockIdx.x * GBN;
  (void)M;

  v8f acc[4] = {};

  for (int k0 = 0; k0 < K; k0 += GBK) {
    // ---- stage A tile 128x16 as float4 (b128 both sides) ----
#pragma unroll
    for (int i = 0; i < 2; ++i) {
      int lin = (t + i * 256) * 4;
      int r = lin >> 4, c = lin & 15;
      float4 v = *(const float4*)(A + (size_t)(mblk + r) * K + k0 + c);
      *(float4*)&lda[r * LDA_STR + c] = v;
    }
    // ---- stage B tile 16x64, k-pair interleaved ----
#pragma unroll
    for (int i = 0; i < 2; ++i) {
      int p  = t + i * 256;                // 0..511 pair-slots
      int pr = p >> 6, c = p & 63;
      v2f pv;
      pv.x = B[(size_t)(k0 + 2 * pr) * N + nblk + c];
      pv.y = B[(size_t)(k0 + 2 * pr + 1) * N + nblk + c];
      *(v2f*)&ldb[pr * LDB_PSTR + c * 2] = pv;
    }
    if (k0 + GBK < K) {                    // gfx1250 global_prefetch of next tiles
      __builtin_prefetch(A + (size_t)(mblk + (t >> 1)) * K + k0 + GBK, 0, 0);
      __builtin_prefetch(B + (size_t)(k0 + GBK + (t & 15)) * N + nblk + (t >> 4) * 4, 0, 0);
    }
    __syncthreads();

    // ---- batch all fragment loads, then a clean WMMA clause ----
    const float* ar = &lda[(wave * 16 + lh) * LDA_STR];
    v2f afr[4], bfr[4][4];
#pragma unroll
    for (int qk = 0; qk < 4; ++qk)
      afr[qk] = *(const v2f*)&ar[4 * qk + 2 * hs];
#pragma unroll
    for (int qk = 0; qk < 4; ++qk)
#pragma unroll
      for (int nt = 0; nt < 4; ++nt)
        bfr[qk][nt] = *(const v2f*)&ldb[(2 * qk + hs) * LDB_PSTR + (nt * 16 + lh) * 2];
#pragma unroll
    for (int qk = 0; qk < 4; ++qk)
#pragma unroll
      for (int nt = 0; nt < 4; ++nt)
        acc[nt] = wmma4(afr[qk], bfr[qk][nt], acc[nt]);
    __syncthreads();
  }

#pragma unroll
  for (int nt = 0; nt < 4; ++nt)
#pragma unroll
    for (int r = 0; r < 8; ++r) {          // C/D layout: row = hs*8+r
      int row = mblk + wave * 16 + hs * 8 + r;
      int col = nblk + nt * 16 + lh;
      C[(size_t)row * N + col] = acc[nt][r];
    }
}

// ---------------------------------------------------------------------------
// Row RMSNorm: one workgroup per row; optional strided second output (cache).
// ---------------------------------------------------------------------------
__global__ __launch_bounds__(256) void rmsnorm_k(
    const float* __restrict__ X, const float* __restrict__ w,
    float* __restrict__ Y, float* __restrict__ Y2,
    int L, int inStride, int outStride, int out2Stride)
{
  __shared__ float red[256];
  const int row = blockIdx.x;
  const int tid = threadIdx.x;
  const float* xr = X + (size_t)row * inStride;

  float ss = 0.f;
  for (int i = tid; i < L; i += 256) { float v = xr[i]; ss += v * v; }
  red[tid] = ss;
  __syncthreads();
  for (int s = 128; s > 0; s >>= 1) {
    if (tid < s) red[tid] += red[tid + s];
    __syncthreads();
  }
  const float scale = rsqrtf(red[0] / (float)L + RMS_EPS);
  for (int i = tid; i < L; i += 256) {
    float v = xr[i] * scale * w[i];
    Y[(size_t)row * outStride + i] = v;
    if (Y2) Y2[(size_t)row * out2Stride + i] = v;
  }
}

// ---------------------------------------------------------------------------
// RoPE: in-place on q_pe (per head), k_pe -> kpe buffer + kv_cache[:,512:].
//   out[d]    = x[2d]c - x[2d+1]s ;  out[d+32] = x[2d+1]c + x[2d]s
// ---------------------------------------------------------------------------
__global__ __launch_bounds__(256) void rope_k(
    float* __restrict__ q, const float* __restrict__ ckv,
    const float* __restrict__ cosT, const float* __restrict__ sinT,
    const int* __restrict__ pos_ids,
    float* __restrict__ kpe, float* __restrict__ cache)
{
  const int row = blockIdx.x;
  const int t   = threadIdx.x;
  const int pos = pos_ids[row];
  const float* cr = cosT + (size_t)pos * D_ROPE;
  const float* sr = sinT + (size_t)pos * D_ROPE;

  float o0[2], o1[2];
  int hh[2], dd[2];
#pragma unroll
  for (int j = 0; j < 2; ++j) {            // 512 q_pe pairs / 256 threads
    int i = t + j * 256;
    int h = i >> 5, d = i & 31;
    const float* base = q + (size_t)row * QW_D + h * Q_HEAD + D_NOPE;
    float x0 = base[2 * d], x1 = base[2 * d + 1];
    float c = cr[d], s = sr[d];
    o0[j] = x0 * c - x1 * s;
    o1[j] = x1 * c + x0 * s;
    hh[j] = h; dd[j] = d;
  }
  __syncthreads();                         // all reads before in-place writes
#pragma unroll
  for (int j = 0; j < 2; ++j) {
    float* base = q + (size_t)row * QW_D + hh[j] * Q_HEAD + D_NOPE;
    base[dd[j]]      = o0[j];
    base[dd[j] + 32] = o1[j];
  }

  if (t < 32) {                            // k_pe (shared across heads)
    int d = t;
    const float* kb = ckv + (size_t)row * CKV_W + KVL_D;
    float x0 = kb[2 * d], x1 = kb[2 * d + 1];
    float c = cr[d], s = sr[d];
    float r0 = x0 * c - x1 * s;
    float r1 = x1 * c + x0 * s;
    kpe[(size_t)row * D_ROPE + d]      = r0;
    kpe[(size_t)row * D_ROPE + d + 32] = r1;
    cache[(size_t)row * CKV_W + KVL_D + d]      = r0;
    cache[(size_t)row * CKV_W + KVL_D + d + 32] = r1;
  }
}

// ---------------------------------------------------------------------------
// Causal attention, flash-style, fp32 WMMA.
// Grid: (S/128, H).  8 waves, each owns 16 query rows.  Key tiles of 32.
// Q fragments in registers (A-layout).  P relayout D->A via per-wave LDS.
// LDS layouts (all fragment reads are aligned b64, conflict-free strides):
//   ldk: row-major stride 196 ; ldv: k-pair interleaved, pair stride 288
//   ldp: per-wave, stride 34
// ---------------------------------------------------------------------------
#define KSTR  196
#define VPSTR 288
#define PSTR  34

__global__ __launch_bounds__(256) void mla_attn_k(
    const float* __restrict__ q, const float* __restrict__ kv,
    const float* __restrict__ kpe, float* __restrict__ out)
{
  __shared__ float ldk[32 * KSTR];          // 24.5 KB  K tile (32 x 192)
  __shared__ float ldv[16 * VPSTR];         // 18.0 KB  V tile (16 pairs x 128 x 2)
  __shared__ float ldp[8 * 16 * PSTR];      // 17.0 KB  per-wave P scratch

  const int t    = threadIdx.x;
  const int wave = t >> 5;
  const int lane = t & 31;
  const int lh   = lane & 15;
  const int hs   = lane >> 4;
  const int h    = blockIdx.y;
  const int qblk = blockIdx.x * 128;
  const int qbase = qblk + wave * 16;
  float* myp = ldp + wave * 16 * PSTR;

  // Q fragments: 48 k-steps of 4 along the 192-dim, A-layout per lane.
  v2f qa[48];
  {
    const float* qp = q + (size_t)(qbase + lh) * QW_D + h * Q_HEAD + 2 * hs;
#pragma unroll
    for (int i = 0; i < 48; ++i) qa[i] = *(const v2f*)(qp + 4 * i);
  }

  v8f acc[8] = {};                          // 16 rows x 128 v-dims
  float mrow[8], lrow[8];
#pragma unroll
  for (int r = 0; r < 8; ++r) { mrow[r] = -1e30f; lrow[r] = 0.f; }

  const int nkb = (qblk >> 5) + 4;          // causal span in 32-key blocks
  for (int kb = 0; kb < nkb; ++kb) {
    const int kbase = kb * 32;

    // ---- stage K tile: k_nope (b128) + k_pe (b128) ----
#pragma unroll
    for (int i = 0; i < 4; ++i) {
      int lin = (t + i * 256) * 4;
      int r = lin >> 7, c = lin & 127;
      float4 v = *(const float4*)(kv + (size_t)(kbase + r) * KV_W + h * 256 + c);
      *(float4*)&ldk[r * KSTR + c] = v;
    }
#pragma unroll
    for (int i = 0; i < 2; ++i) {
      int lin = (t + i * 256) * 4;
      int r = lin >> 6, c = lin & 63;
      float4 v = *(const float4*)(kpe + (size_t)(kbase + r) * D_ROPE + c);
      *(float4*)&ldk[r * KSTR + 128 + c] = v;
    }
    // ---- stage V tile, k-pair interleaved ----
#pragma unroll
    for (int i = 0; i < 8; ++i) {
      int p  = t + i * 256;                 // 0..2047 pair-slots
      int pr = p >> 7, c = p & 127;
      v2f pv;
      pv.x = kv[(size_t)(kbase + 2 * pr) * KV_W + h * 256 + D_NOPE + c];
      pv.y = kv[(size_t)(kbase + 2 * pr + 1) * KV_W + h * 256 + D_NOPE + c];
      *(v2f*)&ldv[pr * VPSTR + c * 2] = pv;
    }
    __syncthreads();

    // ---- S = Q @ K^T : chunk 4 k-steps of loads, then 8 WMMAs ----
    v8f s0 = {}, s1 = {};
#pragma unroll
    for (int kc = 0; kc < 12; ++kc) {
      v2f b0[4], b1[4];
#pragma unroll
      for (int j = 0; j < 4; ++j) {
        int kd = 4 * (4 * kc + j) + 2 * hs;
        b0[j] = *(const v2f*)&ldk[lh * KSTR + kd];
        b1[j] = *(const v2f*)&ldk[(16 + lh) * KSTR + kd];
      }
#pragma unroll
      for (int j = 0; j < 4; ++j) {
        s0 = wmma4(qa[4 * kc + j], b0[j], s0);
        s1 = wmma4(qa[4 * kc + j], b1[j], s1);
      }
    }

    // ---- online softmax (row stats per half-wave lane group) ----
#pragma unroll
    for (int r = 0; r < 8; ++r) {
      int qrow = qbase + hs * 8 + r;
      int key0 = kbase + lh, key1 = kbase + 16 + lh;
      float v0 = (key0 <= qrow) ? s0[r] * ATT_SCALE : -1e9f;
      float v1 = (key1 <= qrow) ? s1[r] * ATT_SCALE : -1e9f;
      float m = fmaxf(v0, v1);
#pragma unroll
      for (int off = 8; off >= 1; off >>= 1) m = fmaxf(m, __shfl_xor(m, off, 32));
      float mnew = fmaxf(mrow[r], m);
      float corr = __expf(mrow[r] - mnew);
      float p0 = __expf(v0 - mnew);
      float p1 = __expf(v1 - mnew);
      float rs = p0 + p1;
#pragma unroll
      for (int off = 8; off >= 1; off >>= 1) rs += __shfl_xor(rs, off, 32);
      lrow[r] = lrow[r] * corr + rs;
      mrow[r] = mnew;
#pragma unroll
      for (int nt = 0; nt < 8; ++nt) acc[nt][r] *= corr;
      int mi = hs * 8 + r;                  // D-layout -> LDS (A-layout source)
      myp[mi * PSTR + lh]      = p0;
      myp[mi * PSTR + 16 + lh] = p1;
    }

    // ---- O += P @ V : chunk 2 k-steps of loads, then 16 WMMAs ----
#pragma unroll
    for (int kc = 0; kc < 4; ++kc) {
      v2f a2[2], bv[2][8];
#pragma unroll
      for (int j = 0; j < 2; ++j) {
        int k4 = 2 * kc + j;
        int kd = 4 * k4 + 2 * hs;
        a2[j] = *(const v2f*)&myp[lh * PSTR + kd];
#pragma unroll
        for (int nt = 0; nt < 8; ++nt)
          bv[j][nt] = *(const v2f*)&ldv[(2 * k4 + hs) * VPSTR + (nt * 16 + lh) * 2];
      }
#pragma unroll
      for (int j = 0; j < 2; ++j)
#pragma unroll
        for (int nt = 0; nt < 8; ++nt)
          acc[nt] = wmma4(a2[j], bv[j][nt], acc[nt]);
    }
    __syncthreads();                        // protect ldk/ldv for next tile
  }

#pragma unroll
  for (int r = 0; r < 8; ++r) {
    float inv = 1.0f / lrow[r];
    int row = qbase + hs * 8 + r;
#pragma unroll
    for (int nt = 0; nt < 8; ++nt)
      out[(size_t)row * (NH * D_V) + h * D_V + nt * 16 + lh] = acc[nt][r] * inv;
  }
}

// ---------------------------------------------------------------------------
// Launch: GEMMs + norms + rope + attention + output projection on `stream`.
// ---------------------------------------------------------------------------
extern "C" void kernel_launch(void* const* d_in, const int* in_sizes, int n_in,
                              void* d_out, int out_size, void* d_ws, size_t ws_size,
                              hipStream_t stream) {
  (void)in_sizes; (void)n_in; (void)out_size; (void)ws_size;
  const float* x       = (const float*)d_in[0];
  /* d_in[1] = mask: causality applied analytically */
  const int*   pos     = (const int*)  d_in[2];
  const float* cosT    = (const float*)d_in[3];
  const float* sinT    = (const float*)d_in[4];
  const float* q_a_w   = (const float*)d_in[5];
  const float* q_a_ln  = (const float*)d_in[6];
  const float* q_b_w   = (const float*)d_in[7];
  const float* kv_a_w  = (const float*)d_in[8];
  const float* kv_a_ln = (const float*)d_in[9];
  const float* kv_b_w  = (const float*)d_in[10];
  const float* o_w     = (const float*)d_in[11];

  float* outp  = (float*)d_out;                       // [S, HID]
  float* cache = outp + (size_t)S_LEN * HID_D;        // [S, 576]

  float* ws    = (float*)d_ws;
  float* q_a   = ws;                                  // [S, 1536]
  float* qbuf  = q_a   + (size_t)S_LEN * QL_D;        // [S, 3072]
  float* ckv   = qbuf  + (size_t)S_LEN * QW_D;        // [S, 576]
  float* cnorm = ckv   + (size_t)S_LEN * CKV_W;       // [S, 512]
  float* kvbuf = cnorm + (size_t)S_LEN * KVL_D;       // [S, 4096]
  float* kpe   = kvbuf + (size_t)S_LEN * KV_W;        // [S, 64]
  float* aout  = kpe   + (size_t)S_LEN * D_ROPE;      // [S, 2048]

  dim3 blk(256);
  // 1) q_a = x @ q_a_w
  gemm_f32_wmma<<<dim3(QL_D / GBN, S_LEN / GBM), blk, 0, stream>>>(
      x, q_a_w, q_a, S_LEN, QL_D, HID_D);
  // 2) rmsnorm(q_a) in place
  rmsnorm_k<<<S_LEN, blk, 0, stream>>>(q_a, q_a_ln, q_a, nullptr,
                                       QL_D, QL_D, QL_D, 0);
  // 3) q = q_a @ q_b_w
  gemm_f32_wmma<<<dim3(QW_D / GBN, S_LEN / GBM), blk, 0, stream>>>(
      q_a, q_b_w, qbuf, S_LEN, QW_D, QL_D);
  // 4) ckv = x @ kv_a_w
  gemm_f32_wmma<<<dim3(CKV_W / GBN, S_LEN / GBM), blk, 0, stream>>>(
      x, kv_a_w, ckv, S_LEN, CKV_W, HID_D);
  // 5) c_norm = rmsnorm(ckv[:, :512]); also -> kv_cache[:, :512]
  rmsnorm_k<<<S_LEN, blk, 0, stream>>>(ckv, kv_a_ln, cnorm, cache,
                                       KVL_D, CKV_W, KVL_D, CKV_W);
  // 6) kv = c_norm @ kv_b_w
  gemm_f32_wmma<<<dim3(KV_W / GBN, S_LEN / GBM), blk, 0, stream>>>(
      cnorm, kv_b_w, kvbuf, S_LEN, KV_W, KVL_D);
  // 7) RoPE q_pe (in place) + k_pe -> kpe + kv_cache[:, 512:]
  rope_k<<<S_LEN, blk, 0, stream>>>(qbuf, ckv, cosT, sinT, pos, kpe, cache);
  // 8) causal attention
  mla_attn_k<<<dim3(S_LEN / 128, NH), blk, 0, stream>>>(qbuf, kvbuf, kpe, aout);
  // 9) out = attn @ o_w
  gemm_f32_wmma<<<dim3(HID_D / GBN, S_LEN / GBM), blk, 0, stream>>>(
      aout, o_w, outp, S_LEN, HID_D, NH * D_V);
}